// GNNModel_18597208392115
// MI455X (gfx1250) — compile-verified
//
#include <hip/hip_runtime.h>
#include <math.h>

typedef __attribute__((ext_vector_type(2))) float v2f;
typedef __attribute__((ext_vector_type(8))) float v8f;

#define NEG_SLOPE 0.2f

__device__ __forceinline__ void atomicMaxFloat(float* addr, float val) {
  // monotone int mapping for IEEE floats
  if (val >= 0.0f) {
    atomicMax((int*)addr, __float_as_int(val));
  } else {
    atomicMin((unsigned int*)addr, (unsigned int)__float_as_int(val));
  }
}

__global__ __launch_bounds__(256) void fill_kernel(float* __restrict__ p, float v, int n) {
  int i = blockIdx.x * 256 + threadIdx.x;
  if (i < n) p[i] = v;
}

// Hout[nrows,128] = X[nrows,128] @ W[128,128]   (row-major, fp32, WMMA f32 16x16x4)
// block = 256 threads = 8 waves; block -> 16-row stripe; wave w -> columns [16w,16w+16)
__global__ __launch_bounds__(256) void gemm128_wmma(const float* __restrict__ X,
                                                    const float* __restrict__ W,
                                                    float* __restrict__ Hout,
                                                    int nrows) {
  __shared__ float tileA[16 * 132];  // 16 rows x 128 cols, +4 pad vs bank conflicts
  const int tid  = threadIdx.x;
  const int wave = tid >> 5;
  const int lane = tid & 31;
  const int row0 = blockIdx.x << 4;

  // stage A tile (coalesced float4 loads)
  for (int i = tid; i < 512; i += 256) {
    int r   = i >> 5;   // 32 float4 per row
    int c4  = i & 31;
    int row = row0 + r;
    if (row >= nrows) row = nrows - 1;
    const float4 v = ((const float4*)(X + (size_t)row * 128))[c4];
    float* dst = &tileA[r * 132 + (c4 << 2)];
    dst[0] = v.x; dst[1] = v.y; dst[2] = v.z; dst[3] = v.w;
  }
  __syncthreads();

  const int col0 = wave << 4;
  const int half = lane >> 4;   // 0: K pair (k0,k0+1) ; 1: (k0+2,k0+3)
  const int lid  = lane & 15;   // M for A-frag, N for B-frag
  v8f acc = {};
#pragma unroll 8
  for (int k0 = 0; k0 < 128; k0 += 4) {
    const int ka = k0 + (half << 1);
    v2f a, b;
    a.x = tileA[lid * 132 + ka];
    a.y = tileA[lid * 132 + ka + 1];
    b.x = W[(size_t)ka * 128 + col0 + lid];
    b.y = W[(size_t)(ka + 1) * 128 + col0 + lid];
    acc = __builtin_amdgcn_wmma_f32_16x16x4_f32(false, a, false, b, (short)0, acc,
                                                false, false);
  }
  // C/D layout: VGPR v -> row (v + 8*half), col = lid
  float* outp = Hout + (size_t)(row0 + (half << 3)) * 128 + col0 + lid;
  if (row0 + 16 <= nrows) {
    // full tile: branchless stores (common case; N % 16 == 0 here)
#pragma unroll
    for (int v = 0; v < 8; ++v) outp[(size_t)v * 128] = acc[v];
  } else {
#pragma unroll
    for (int v = 0; v < 8; ++v) {
      const int r = row0 + v + (half << 3);
      if (r < nrows) Hout[(size_t)r * 128 + col0 + lid] = acc[v];
    }
  }
}

// per-node attention coefficients: alpha_s = h . a_src, alpha_d = h . a_dst
__global__ __launch_bounds__(256) void node_alpha(const float* __restrict__ Hf,
                                                  const float* __restrict__ a_src,
                                                  const float* __restrict__ a_dst,
                                                  float* __restrict__ as,
                                                  float* __restrict__ ad, int n) {
  int node = (blockIdx.x * 256 + threadIdx.x) >> 5;
  int lane = threadIdx.x & 31;
  if (node >= n) return;
  const float4 hv = ((const float4*)(Hf + (size_t)node * 128))[lane];
  const float4 s  = ((const float4*)a_src)[lane];
  const float4 d  = ((const float4*)a_dst)[lane];
  float vs = hv.x * s.x + hv.y * s.y + hv.z * s.z + hv.w * s.w;
  float vd = hv.x * d.x + hv.y * d.y + hv.z * d.z + hv.w * d.w;
#pragma unroll
  for (int off = 16; off > 0; off >>= 1) {
    vs += __shfl_xor(vs, off, 32);
    vd += __shfl_xor(vd, off, 32);
  }
  if (lane == 0) { as[node] = vs; ad[node] = vd; }
}

__device__ __forceinline__ void edge_sd(const int* __restrict__ ei, int i, int E,
                                        int& s, int& d) {
  if (i < E) { s = ei[i]; d = ei[E + i]; }
  else       { s = i - E; d = i - E; }   // self-loops appended
}

__global__ __launch_bounds__(256) void edge_max(const int* __restrict__ ei,
                                                const float* __restrict__ as,
                                                const float* __restrict__ ad,
                                                float* __restrict__ emax, int E, int Etot) {
  int i = blockIdx.x * 256 + threadIdx.x;
  if (i >= Etot) return;
  int s, d; edge_sd(ei, i, E, s, d);
  float e = as[s] + ad[d];
  e = e > 0.f ? e : NEG_SLOPE * e;
  atomicMaxFloat(&emax[d], e);
}

__global__ __launch_bounds__(256) void edge_exp(const int* __restrict__ ei,
                                                const float* __restrict__ as,
                                                const float* __restrict__ ad,
                                                const float* __restrict__ emax,
                                                float* __restrict__ denom,
                                                float* __restrict__ eebuf, int E, int Etot) {
  int i = blockIdx.x * 256 + threadIdx.x;
  if (i >= Etot) return;
  int s, d; edge_sd(ei, i, E, s, d);
  float e = as[s] + ad[d];
  e = e > 0.f ? e : NEG_SLOPE * e;
  float ee = __expf(e - emax[d]);
  eebuf[i] = ee;
  atomicAdd(&denom[d], ee);
}

// one wave per edge: Out[dst] += alpha * H[src]  (coalesced 512B gather per edge)
__global__ __launch_bounds__(256) void edge_msg(const int* __restrict__ ei,
                                                const float* __restrict__ eebuf,
                                                const float* __restrict__ denom,
                                                const float* __restrict__ Hf,
                                                float* __restrict__ Out, int E, int Etot) {
  int w = (blockIdx.x * 256 + threadIdx.x) >> 5;
  int lane = threadIdx.x & 31;
  if (w >= Etot) return;
  int s, d; edge_sd(ei, w, E, s, d);
  float alpha = eebuf[w] / (denom[d] + 1e-16f);
  const float4 hv = ((const float4*)(Hf + (size_t)s * 128))[lane];
  float* o = Out + (size_t)d * 128 + (lane << 2);
  atomicAdd(o + 0, hv.x * alpha);
  atomicAdd(o + 1, hv.y * alpha);
  atomicAdd(o + 2, hv.z * alpha);
  atomicAdd(o + 3, hv.w * alpha);
}

__global__ __launch_bounds__(256) void bias_relu(float* __restrict__ O,
                                                 const float* __restrict__ b, int n128) {
  int i = blockIdx.x * 256 + threadIdx.x;
  if (i >= n128) return;
  float v = O[i] + b[i & 127];
  O[i] = v > 0.f ? v : 0.f;
}

__global__ __launch_bounds__(256) void pool_sum(const float* __restrict__ O,
                                                const int* __restrict__ batch,
                                                float* __restrict__ pooled,
                                                float* __restrict__ counts, int n) {
  int node = (blockIdx.x * 256 + threadIdx.x) >> 5;
  int lane = threadIdx.x & 31;
  if (node >= n) return;
  int g = batch[node];
  const float4 hv = ((const float4*)(O + (size_t)node * 128))[lane];
  float* p = pooled + (size_t)g * 128 + (lane << 2);
  atomicAdd(p + 0, hv.x);
  atomicAdd(p + 1, hv.y);
  atomicAdd(p + 2, hv.z);
  atomicAdd(p + 3, hv.w);
  if (lane == 0) atomicAdd(&counts[g], 1.0f);
}

// out[g, c] = (pooled[g]/count[g]) @ Wc[:, c] + bc[c]   (one wave per graph)
__global__ __launch_bounds__(256) void classify(const float* __restrict__ pooled,
                                                const float* __restrict__ counts,
                                                const float* __restrict__ Wc,
                                                const float* __restrict__ bc,
                                                float* __restrict__ out, int G, int C) {
  int g = (blockIdx.x * 256 + threadIdx.x) >> 5;
  int lane = threadIdx.x & 31;
  if (g >= G) return;
  float inv = 1.0f / fmaxf(counts[g], 1.0f);
  float4 pv = ((const float4*)(pooled + (size_t)g * 128))[lane];
  pv.x *= inv; pv.y *= inv; pv.z *= inv; pv.w *= inv;
  int f0 = lane << 2;
  for (int c = 0; c < C; ++c) {
    float acc = pv.x * Wc[(f0 + 0) * C + c] + pv.y * Wc[(f0 + 1) * C + c] +
                pv.z * Wc[(f0 + 2) * C + c] + pv.w * Wc[(f0 + 3) * C + c];
#pragma unroll
    for (int off = 16; off > 0; off >>= 1) acc += __shfl_xor(acc, off, 32);
    if (lane == 0) out[g * C + c] = acc + bc[c];
  }
}

extern "C" void kernel_launch(void* const* d_in, const int* in_sizes, int n_in,
                              void* d_out, int out_size, void* d_ws, size_t ws_size,
                              hipStream_t stream) {
  const float* x    = (const float*)d_in[0];
  const int*   ei   = (const int*)d_in[1];
  const int*   batch= (const int*)d_in[2];
  const float* W1   = (const float*)d_in[4];
  const float* as1  = (const float*)d_in[5];
  const float* ad1  = (const float*)d_in[6];
  const float* b1   = (const float*)d_in[7];
  const float* W2   = (const float*)d_in[8];
  const float* as2  = (const float*)d_in[9];
  const float* ad2  = (const float*)d_in[10];
  const float* b2   = (const float*)d_in[11];
  const float* Wc   = (const float*)d_in[12];
  const float* bc   = (const float*)d_in[13];
  float* out = (float*)d_out;
  (void)n_in; (void)ws_size;

  const int N    = in_sizes[2];
  const int E    = in_sizes[1] / 2;
  const int Etot = E + N;
  const int C    = in_sizes[13];
  const int G    = out_size / C;

  size_t off = 0;
  auto wsalloc = [&](size_t bytes) -> float* {
    float* p = (float*)((char*)d_ws + off);
    off += (bytes + 255) & ~(size_t)255;
    return p;
  };
  float* Hbuf   = wsalloc((size_t)N * 128 * 4);
  float* O1     = wsalloc((size_t)N * 128 * 4);
  float* O2     = wsalloc((size_t)N * 128 * 4);
  float* alphaS = wsalloc((size_t)N * 4);
  float* alphaD = wsalloc((size_t)N * 4);
  float* emax   = wsalloc((size_t)N * 4);
  float* denom  = wsalloc((size_t)N * 4);
  float* eebuf  = wsalloc((size_t)Etot * 4);
  float* pooled = wsalloc((size_t)G * 128 * 4);
  float* counts = wsalloc((size_t)G * 4);

  auto cdiv = [](int a, int b) { return (a + b - 1) / b; };
  dim3 blk(256);

  const float* layer_in[2] = {x, O1};
  const float* Ws[2]  = {W1, W2};
  const float* asv[2] = {as1, as2};
  const float* adv[2] = {ad1, ad2};
  const float* bv[2]  = {b1, b2};
  float* Ov[2]        = {O1, O2};

  for (int L = 0; L < 2; ++L) {
    fill_kernel<<<cdiv(N, 256), blk, 0, stream>>>(emax, -INFINITY, N);
    fill_kernel<<<cdiv(N, 256), blk, 0, stream>>>(denom, 0.0f, N);
    fill_kernel<<<cdiv(N * 128, 256), blk, 0, stream>>>(Ov[L], 0.0f, N * 128);
    gemm128_wmma<<<cdiv(N, 16), blk, 0, stream>>>(layer_in[L], Ws[L], Hbuf, N);
    node_alpha<<<cdiv(N, 8), blk, 0, stream>>>(Hbuf, asv[L], adv[L], alphaS, alphaD, N);
    edge_max<<<cdiv(Etot, 256), blk, 0, stream>>>(ei, alphaS, alphaD, emax, E, Etot);
    edge_exp<<<cdiv(Etot, 256), blk, 0, stream>>>(ei, alphaS, alphaD, emax, denom, eebuf, E, Etot);
    edge_msg<<<cdiv(Etot, 8), blk, 0, stream>>>(ei, eebuf, denom, Hbuf, Ov[L], E, Etot);
    bias_relu<<<cdiv(N * 128, 256), blk, 0, stream>>>(Ov[L], bv[L], N * 128);
  }
  fill_kernel<<<cdiv(G * 128, 256), blk, 0, stream>>>(pooled, 0.0f, G * 128);
  fill_kernel<<<cdiv(G, 256), blk, 0, stream>>>(counts, 0.0f, G);
  pool_sum<<<cdiv(N, 8), blk, 0, stream>>>(O2, batch, pooled, counts, N);
  classify<<<cdiv(G, 8), blk, 0, stream>>>(pooled, counts, Wc, bc, out, G, C);
}